// Gate_21577915695170
// MI455X (gfx1250) — compile-verified
//
#include <hip/hip_runtime.h>

typedef __attribute__((ext_vector_type(2))) float v2f;
typedef __attribute__((ext_vector_type(4))) float v4f;
typedef __attribute__((ext_vector_type(8))) float v8f;

#define IN_DIM   256
#define NUM_EXP  64
#define WAVES    2
#define BLOCK    (WAVES * 32)
#define TPW      16                  // tokens per wave (one 16x16 WMMA tile)
#define TPB      (WAVES * TPW)

// LDS layout (float offsets). Pads chosen for bank-conflict-free frag reads.
#define XPAD   260                   // 256 + 4   (260 mod 64 == 4)
#define XWSZ   (16 * XPAD)           // 4160 floats per-wave x tile
#define W1OFF  (WAVES * XWSZ)        // 8320
#define W1PAD  260
#define W2OFF  (W1OFF + 16 * W1PAD)  // 12480
#define W2PAD  20
#define LDSSZ  (W2OFF + NUM_EXP * W2PAD) // 13760 floats = 55040 bytes
#define HPAD   20                    // h tile 16x16 padded, lives inside x region
#define PPAD   68                    // logits/probs 16x64 padded, inside x region
#define POFF   1024                  // pbuf offset within wave x region

// Cross-half-wave exchange: ds_swizzle group-of-32, xor_mask=0x10, and=0x1f
__device__ __forceinline__ float swap16(float v) {
    return __int_as_float(__builtin_amdgcn_ds_swizzle(__float_as_int(v), 0x401F));
}

// Tree max over 32 floats held in (fully unrolled) registers
__device__ __forceinline__ float max32_tree(const float* w) {
    float a[16];
#pragma unroll
    for (int i = 0; i < 16; ++i) a[i] = fmaxf(w[i], w[i + 16]);
#pragma unroll
    for (int i = 0; i < 8; ++i)  a[i] = fmaxf(a[i], a[i + 8]);
#pragma unroll
    for (int i = 0; i < 4; ++i)  a[i] = fmaxf(a[i], a[i + 4]);
    return fmaxf(fmaxf(a[0], a[2]), fmaxf(a[1], a[3]));
}

__global__ __launch_bounds__(BLOCK)
void moe_gate_topk_kernel(const float* __restrict__ x,
                          const float* __restrict__ W1,
                          const float* __restrict__ b1,
                          const float* __restrict__ W2,
                          const float* __restrict__ b2,
                          float* __restrict__ out)
{
    __shared__ float lds[LDSSZ];
    const int tid  = threadIdx.x;
    const int wv   = tid >> 5;
    const int lane = tid & 31;
    const int hl   = lane >> 4;      // lane half (0/1) -> K sub-pair in frags
    const int l16  = lane & 15;

    const long long waveTok = (long long)blockIdx.x * TPB + (long long)wv * TPW;

    float* xw  = lds + wv * XWSZ;
    float* w1s = lds + W1OFF;
    float* w2s = lds + W2OFF;

    // ---------------- stage W1 (16x256 -> pad 260), coalesced b128 ----------
#pragma unroll
    for (int it = 0; it < (16 * IN_DIM) / 4 / BLOCK; ++it) {
        int i = tid + it * BLOCK;
        v4f v = *(const v4f*)(W1 + i * 4);
        int r = i >> 6;              // 64 float4 per row
        int c = (i & 63) << 2;
        *(v4f*)(w1s + r * W1PAD + c) = v;
    }
    // ---------------- stage W2 (64x16 -> pad 20) ----------------------------
#pragma unroll
    for (int it = 0; it < (NUM_EXP * 16) / 4 / BLOCK; ++it) {
        int i = tid + it * BLOCK;
        v4f v = *(const v4f*)(W2 + i * 4);
        int r = i >> 2;              // 4 float4 per row
        int c = (i & 3) << 2;
        *(v4f*)(w2s + r * W2PAD + c) = v;
    }
    // ---------------- stage this wave's 16x256 token tile (streaming) -------
    const float* xg = x + waveTok * IN_DIM;
#pragma unroll
    for (int it = 0; it < (TPW * IN_DIM) / 4 / 32; ++it) {
        int i = lane + it * 32;
        v4f v = __builtin_nontemporal_load((const v4f*)(xg + i * 4));
        int r = i >> 6;
        int c = (i & 63) << 2;
        *(v4f*)(xw + r * XPAD + c) = v;
    }
    __syncthreads();

    // ---------------- fc1: h = relu(x @ W1^T + b1) via WMMA f32 16x16x4 -----
    // A frag (16x4 f32): lane holds x[row=l16][kk + 2*hl + {0,1}]
    // B frag (4x16 f32): lane holds W1[col=l16][kk + 2*hl + {0,1}]  (B[k][n]=W1[n][k])
    // C/D  : lane l, vgpr v -> [M = v + 8*hl][N = l16]; bias b1[N] broadcast per lane
    const float bb1 = b1[l16];
    v8f acc;
#pragma unroll
    for (int c = 0; c < 8; ++c) acc[c] = bb1;

    const int aoff = l16 * XPAD  + 2 * hl;
    const int boff = l16 * W1PAD + 2 * hl;
#pragma unroll
    for (int kk = 0; kk < IN_DIM; kk += 4) {
        v2f a = *(const v2f*)(xw  + aoff + kk);
        v2f b = *(const v2f*)(w1s + boff + kk);
        acc = __builtin_amdgcn_wmma_f32_16x16x4_f32(
                  false, a, false, b, (short)0, acc, false, false);
    }
#pragma unroll
    for (int c = 0; c < 8; ++c) acc[c] = fmaxf(acc[c], 0.0f);

    // D-layout -> A-layout transpose of h through LDS (reuse dead x region)
    float* hb = xw;                  // 16 x HPAD
#pragma unroll
    for (int c = 0; c < 8; ++c) {
        int M = c + 8 * hl;
        hb[M * HPAD + l16] = acc[c];
    }
    __syncthreads();

    // ---------------- fc2: logits = h @ W2^T + b2 ---------------------------
    v8f lacc[4];
#pragma unroll
    for (int tn = 0; tn < 4; ++tn) {
        const float bb2 = b2[tn * 16 + l16];
        v8f a2;
#pragma unroll
        for (int c = 0; c < 8; ++c) a2[c] = bb2;
#pragma unroll
        for (int s = 0; s < 4; ++s) {
            v2f a = *(const v2f*)(hb  + l16 * HPAD + 4 * s + 2 * hl);
            v2f b = *(const v2f*)(w2s + (tn * 16 + l16) * W2PAD + 4 * s + 2 * hl);
            a2 = __builtin_amdgcn_wmma_f32_16x16x4_f32(
                     false, a, false, b, (short)0, a2, false, false);
        }
        lacc[tn] = a2;
    }

    // logits (D layout) -> per-token rows in LDS
    float* pb = xw + POFF;           // 16 x PPAD
#pragma unroll
    for (int tn = 0; tn < 4; ++tn)
#pragma unroll
        for (int c = 0; c < 8; ++c) {
            int M = c + 8 * hl;
            pb[M * PPAD + tn * 16 + l16] = lacc[tn][c];
        }
    __syncthreads();

    // ---------------- softmax + top-8 + renormalize -------------------------
    // Token t is co-owned by lane t (experts 0..31) and lane t+16 (experts 32..63).
    // Cross-half reductions via ds_swizzle SWAPX16; whole wave stays active.
    {
        const int t = l16;
        float e[32], w[32];
#pragma unroll
        for (int j = 0; j < 8; ++j) {
            v4f v = *(const v4f*)(pb + t * PPAD + 32 * hl + 4 * j);
            e[4 * j + 0] = v[0]; e[4 * j + 1] = v[1];
            e[4 * j + 2] = v[2]; e[4 * j + 3] = v[3];
        }
        float m = max32_tree(e);
        m = fmaxf(m, swap16(m));                 // global max over 64
        float S = 0.0f;
#pragma unroll
        for (int i = 0; i < 32; ++i) {
            e[i] = __expf(e[i] - m);
            S += e[i];
            w[i] = e[i];
        }
        const float Z = S + swap16(S);           // global partition sum

        // 8 passes: global max via local tree-max + cross-half fmax,
        // only the owning lane's copy equals cur -> masking is local.
        float s8 = 0.0f, thr = 0.0f;
#pragma unroll
        for (int p = 0; p < 8; ++p) {
            float cur = max32_tree(w);
            cur = fmaxf(cur, swap16(cur));
            s8 += cur;
            thr = cur;                            // last pass = 8th largest
#pragma unroll
            for (int i = 0; i < 32; ++i)
                w[i] = (w[i] == cur) ? -1.0f : w[i];
        }
        const float invZ  = 1.0f / Z;
        const float scale = invZ / (s8 * invZ + 1e-12f); // p_i / (sum8_p + eps)
#pragma unroll
        for (int j = 0; j < 8; ++j) {
            v4f o;
#pragma unroll
            for (int q = 0; q < 4; ++q) {
                int i = 4 * j + q;
                o[q] = (e[i] >= thr) ? e[i] * scale : 0.0f;
            }
            *(v4f*)(pb + t * PPAD + 32 * hl + 4 * j) = o;
        }
    }
    __syncthreads();

    // ---------------- coalesced streaming store: 16 tokens x 64 experts -----
    float* og = out + waveTok * NUM_EXP;
#pragma unroll
    for (int i = 0; i < 8; ++i) {
        int f = i * 32 + lane;       // float4 index within wave's 1024 floats
        int t = f >> 4;              // 16 float4 per token row
        int c = (f & 15) << 2;
        v4f v = *(const v4f*)(pb + t * PPAD + c);
        __builtin_nontemporal_store(v, (v4f*)(og + f * 4));
    }
}

extern "C" void kernel_launch(void* const* d_in, const int* in_sizes, int n_in,
                              void* d_out, int out_size, void* d_ws, size_t ws_size,
                              hipStream_t stream) {
    const float* x  = (const float*)d_in[0];
    const float* W1 = (const float*)d_in[1];
    const float* b1 = (const float*)d_in[2];
    const float* W2 = (const float*)d_in[3];
    const float* b2 = (const float*)d_in[4];
    float* out = (float*)d_out;

    const int ntok = in_sizes[0] / IN_DIM;   // 1048576
    dim3 grid(ntok / TPB);                   // 32768 blocks of 2 waves
    moe_gate_topk_kernel<<<grid, BLOCK, 0, stream>>>(x, W1, b1, W2, b2, out);
}